// GCN_730144441188
// MI455X (gfx1250) — compile-verified
//
#include <hip/hip_runtime.h>
#include <math.h>

typedef __attribute__((ext_vector_type(2))) float v2f;
typedef __attribute__((ext_vector_type(8))) float v8f;

#define DIM   128
#define NGR   256     // graphs
#define NCLS  10      // classes
#define NLAY  5
#define BN_EPS 1e-5f

// ---------------------------------------------------------------------------
// degree / norm precompute
// ---------------------------------------------------------------------------
__global__ void k_init_deg(float* __restrict__ deg, int nN) {
    int i = blockIdx.x * blockDim.x + threadIdx.x;
    if (i < nN) deg[i] = 1.0f;                    // self loop contributes 1
}

__global__ void k_count_deg(float* __restrict__ deg, const int* __restrict__ dst, int nE) {
    int e = blockIdx.x * blockDim.x + threadIdx.x;
    if (e < nE) atomicAdd(&deg[dst[e]], 1.0f);
}

__global__ void k_finish_dinv(float* __restrict__ deg, int nN) {
    int i = blockIdx.x * blockDim.x + threadIdx.x;
    if (i < nN) {
        float d = deg[i];
        deg[i] = (d > 0.0f) ? rsqrtf(d) : 0.0f;   // deg >= 1 always, but be safe
    }
}

__global__ void k_edge_norm(float* __restrict__ norm, const float* __restrict__ dinv,
                            const int* __restrict__ src, const int* __restrict__ dst, int nE) {
    int e = blockIdx.x * blockDim.x + threadIdx.x;
    if (e < nE) norm[e] = dinv[src[e]] * dinv[dst[e]];
}

// bnScale[k] = gamma/sqrt(var+eps); bnShift[k] = beta - mean*bnScale
__global__ void k_bn_precompute(const float* __restrict__ gam, const float* __restrict__ bet,
                                const float* __restrict__ mu,  const float* __restrict__ var,
                                float* __restrict__ bnS, float* __restrict__ bnB, int total) {
    int i = blockIdx.x * blockDim.x + threadIdx.x;
    if (i < total) {
        float s = gam[i] * rsqrtf(var[i] + BN_EPS);
        bnS[i] = s;
        bnB[i] = bet[i] - mu[i] * s;
    }
}

// ---------------------------------------------------------------------------
// GEMM: out[N,128] = act(H)[N,128] @ W[128,128]
// act = identity (bnS==nullptr) or relu(x*bnS + bnB)  (fused BN+ReLU of the
// previous layer). One wave32 per 16-row strip, 8x v_wmma_f32_16x16x4_f32
// per K-step, K-loop of 32 steps.
// ---------------------------------------------------------------------------
__global__ void k_gemm_wmma(const float* __restrict__ H, const float* __restrict__ W,
                            const float* __restrict__ bnS, const float* __restrict__ bnB,
                            float* __restrict__ out, int nRows) {
    const int lane  = threadIdx.x & 31;
    const int wave  = threadIdx.x >> 5;
    const int mTile = blockIdx.x * (blockDim.x >> 5) + wave;
    const int row16 = mTile * 16;
    if (row16 >= nRows) return;

    if (row16 + 16 <= nRows) {
        // ---- full-tile WMMA path (EXEC all ones within this wave) ----
        const v8f zero = {0.f, 0.f, 0.f, 0.f, 0.f, 0.f, 0.f, 0.f};
        v8f acc0 = zero, acc1 = zero, acc2 = zero, acc3 = zero;
        v8f acc4 = zero, acc5 = zero, acc6 = zero, acc7 = zero;

        const int half = lane >> 4;          // 0: K=k0..k0+1, 1: K=k0+2..k0+3
        const int l15  = lane & 15;
        const int koff = half * 2;
        const float* arow = H + (size_t)(row16 + l15) * DIM;

        for (int k0 = 0; k0 < DIM; k0 += 4) {
            const int ka = k0 + koff;
            v2f a;
            {
                // 8B-aligned pair load of A row
                const float2 av = *(const float2*)(arow + ka);
                float ax = av.x, ay = av.y;
                if (bnS) {
                    ax = fmaxf(0.f, fmaf(ax, bnS[ka],     bnB[ka]));
                    ay = fmaxf(0.f, fmaf(ay, bnS[ka + 1], bnB[ka + 1]));
                }
                a.x = ax; a.y = ay;
            }
            const float* wr = W + (size_t)ka * DIM + l15;   // row ka, col base l15
#define WMMA_STEP(T, ACC)                                                            \
            {                                                                        \
                v2f b;                                                               \
                b.x = wr[(T) * 16];                                                  \
                b.y = wr[DIM + (T) * 16];                                            \
                ACC = __builtin_amdgcn_wmma_f32_16x16x4_f32(false, a, false, b,      \
                                                            (short)0, ACC, false,   \
                                                            false);                 \
            }
            WMMA_STEP(0, acc0) WMMA_STEP(1, acc1) WMMA_STEP(2, acc2) WMMA_STEP(3, acc3)
            WMMA_STEP(4, acc4) WMMA_STEP(5, acc5) WMMA_STEP(6, acc6) WMMA_STEP(7, acc7)
#undef WMMA_STEP
        }

        // store: VGPR i of C/D holds (M=i, N=lane) for lanes 0-15 and
        // (M=i+8, N=lane-16) for lanes 16-31
        const int srow = row16 + half * 8;
#define STORE_TILE(T, ACC)                                                           \
        {                                                                            \
            _Pragma("unroll")                                                        \
            for (int i = 0; i < 8; ++i)                                              \
                out[(size_t)(srow + i) * DIM + (T) * 16 + l15] = ACC[i];             \
        }
        STORE_TILE(0, acc0) STORE_TILE(1, acc1) STORE_TILE(2, acc2) STORE_TILE(3, acc3)
        STORE_TILE(4, acc4) STORE_TILE(5, acc5) STORE_TILE(6, acc6) STORE_TILE(7, acc7)
#undef STORE_TILE
    } else {
        // ---- scalar tail (never hit for N=100000, kept for safety) ----
        for (int r = row16 + lane; r < nRows; r += 32) {
            for (int c = 0; c < DIM; ++c) {
                float s = 0.f;
                for (int k = 0; k < DIM; ++k) {
                    float v = H[(size_t)r * DIM + k];
                    if (bnS) v = fmaxf(0.f, fmaf(v, bnS[k], bnB[k]));
                    s = fmaf(v, W[(size_t)k * DIM + c], s);
                }
                out[(size_t)r * DIM + c] = s;
            }
        }
    }
}

// ---------------------------------------------------------------------------
// aggregation: out[n,:] = bias + dinv[n]^2 * tmp[n,:]   (self loop + bias)
// then edge kernel: out[dst,:] += norm[e] * tmp[src,:]  (atomics)
// ---------------------------------------------------------------------------
__global__ void k_agg_init(const float* __restrict__ tmp, const float* __restrict__ dinv,
                           const float* __restrict__ bias, float* __restrict__ out, int nN) {
    int i = blockIdx.x * blockDim.x + threadIdx.x;
    int total = nN * (DIM / 4);
    if (i >= total) return;
    int n = i >> 5;                    // DIM/4 == 32 float4 per row
    int q = i & 31;
    float di = dinv[n];
    float sn = di * di;
    const float4 t = ((const float4*)tmp)[(size_t)n * 32 + q];
    const float4 b = ((const float4*)bias)[q];
    float4 o;
    o.x = fmaf(sn, t.x, b.x);
    o.y = fmaf(sn, t.y, b.y);
    o.z = fmaf(sn, t.z, b.z);
    o.w = fmaf(sn, t.w, b.w);
    ((float4*)out)[(size_t)n * 32 + q] = o;
}

__global__ void k_agg_edges(const float* __restrict__ tmp, const int* __restrict__ src,
                            const int* __restrict__ dst, const float* __restrict__ norm,
                            float* __restrict__ out, int nE) {
    const int lane = threadIdx.x & 31;
    const int e = blockIdx.x * (blockDim.x >> 5) + (threadIdx.x >> 5);
    if (e >= nE) return;
    const int s = src[e];
    const int d = dst[e];
    const float w = norm[e];
    const float4 v = ((const float4*)(tmp + (size_t)s * DIM))[lane];
    float* o = out + (size_t)d * DIM + lane * 4;
    atomicAdd(o + 0, v.x * w);
    atomicAdd(o + 1, v.y * w);
    atomicAdd(o + 2, v.z * w);
    atomicAdd(o + 3, v.w * w);
}

// ---------------------------------------------------------------------------
// pooling + classification head
// ---------------------------------------------------------------------------
__global__ void k_zero(float* __restrict__ p, int total) {
    int i = blockIdx.x * blockDim.x + threadIdx.x;
    if (i < total) p[i] = 0.0f;
}

__global__ void k_pool_nodes(const float* __restrict__ h, const int* __restrict__ batch,
                             float* __restrict__ pooled, float* __restrict__ cnt, int nN) {
    const int lane = threadIdx.x & 31;
    const int n = blockIdx.x * (blockDim.x >> 5) + (threadIdx.x >> 5);
    if (n >= nN) return;
    const int g = batch[n];
    const float4 v = ((const float4*)(h + (size_t)n * DIM))[lane];
    float* p = pooled + (size_t)g * DIM + lane * 4;
    atomicAdd(p + 0, v.x);
    atomicAdd(p + 1, v.y);
    atomicAdd(p + 2, v.z);
    atomicAdd(p + 3, v.w);
    if (lane == 0) atomicAdd(&cnt[g], 1.0f);
}

__global__ void k_head(const float* __restrict__ pooled, const float* __restrict__ cnt,
                       const float* __restrict__ outW, const float* __restrict__ outb,
                       float* __restrict__ out) {
    const int g = threadIdx.x;           // one block of NGR threads
    if (g >= NGR) return;
    const float invc = 1.0f / fmaxf(cnt[g], 1.0f);
    float acc[NCLS];
#pragma unroll
    for (int c = 0; c < NCLS; ++c) acc[c] = outb[c];
    for (int d = 0; d < DIM; ++d) {
        const float p = pooled[(size_t)g * DIM + d] * invc;
#pragma unroll
        for (int c = 0; c < NCLS; ++c)
            acc[c] = fmaf(p, outW[d * NCLS + c], acc[c]);
    }
    float m = acc[0];
#pragma unroll
    for (int c = 1; c < NCLS; ++c) m = fmaxf(m, acc[c]);
    float s = 0.f;
#pragma unroll
    for (int c = 0; c < NCLS; ++c) s += expf(acc[c] - m);
    const float ls = logf(s);
#pragma unroll
    for (int c = 0; c < NCLS; ++c) out[g * NCLS + c] = acc[c] - m - ls;
}

// ---------------------------------------------------------------------------
// launch
// ---------------------------------------------------------------------------
extern "C" void kernel_launch(void* const* d_in, const int* in_sizes, int n_in,
                              void* d_out, int out_size, void* d_ws, size_t ws_size,
                              hipStream_t stream) {
    const float* x     = (const float*)d_in[0];
    const int*   src   = (const int*)d_in[1];
    const int*   dst   = (const int*)d_in[2];
    const int*   batch = (const int*)d_in[3];
    const float* Ws    = (const float*)d_in[4];
    const float* bs    = (const float*)d_in[5];
    const float* gam   = (const float*)d_in[6];
    const float* bet   = (const float*)d_in[7];
    const float* mu    = (const float*)d_in[8];
    const float* var   = (const float*)d_in[9];
    const float* outW  = (const float*)d_in[10];
    const float* outb  = (const float*)d_in[11];

    const int nN = in_sizes[3];          // 100000 nodes (batch array length)
    const int nE = in_sizes[1];          // 1.6M edges

    // workspace layout (floats)
    float* ws     = (float*)d_ws;
    float* deg    = ws;                                  // N   (dinv after finish)
    float* norm   = deg + nN;                            // E
    float* bnS    = norm + nE;                           // (L-1)*128
    float* bnB    = bnS + (NLAY - 1) * DIM;              // (L-1)*128
    float* pooled = bnB + (NLAY - 1) * DIM;              // G*128
    float* cnt    = pooled + (size_t)NGR * DIM;          // G
    size_t off    = (size_t)((cnt + NGR) - ws);
    off = (off + 3) & ~(size_t)3;                        // 16B align
    float* bufA   = ws + off;                            // N*128 (gemm tmp)
    float* bufB   = bufA + (size_t)nN * DIM;             // N*128 (node features)

    // --- degree / norm / bn precompute ---
    k_init_deg  <<<(nN + 255) / 256, 256, 0, stream>>>(deg, nN);
    k_count_deg <<<(nE + 255) / 256, 256, 0, stream>>>(deg, dst, nE);
    k_finish_dinv<<<(nN + 255) / 256, 256, 0, stream>>>(deg, nN);
    k_edge_norm <<<(nE + 255) / 256, 256, 0, stream>>>(norm, deg, src, dst, nE);
    k_bn_precompute<<<2, 256, 0, stream>>>(gam, bet, mu, var, bnS, bnB, (NLAY - 1) * DIM);

    // --- 5 GCN layers: tmp = act(h) @ W ; h' = Ahat*tmp + b ---
    const int mTiles     = (nN + 15) / 16;
    const int gemmBlocks = (mTiles + 3) / 4;             // 4 waves / block
    const int initBlocks = (nN * (DIM / 4) + 255) / 256;
    const int edgeBlocks = (nE + 7) / 8;                 // 8 waves / block

    const float* hin = x;
    for (int l = 0; l < NLAY; ++l) {
        const float* bnSl = (l == 0) ? nullptr : bnS + (l - 1) * DIM;
        const float* bnBl = (l == 0) ? nullptr : bnB + (l - 1) * DIM;
        k_gemm_wmma<<<gemmBlocks, 128, 0, stream>>>(hin, Ws + (size_t)l * DIM * DIM,
                                                    bnSl, bnBl, bufA, nN);
        k_agg_init <<<initBlocks, 256, 0, stream>>>(bufA, deg, bs + (size_t)l * DIM, bufB, nN);
        k_agg_edges<<<edgeBlocks, 256, 0, stream>>>(bufA, src, dst, norm, bufB, nE);
        hin = bufB;
    }

    // --- mean pool + linear head + log_softmax ---
    k_zero<<<(NGR * DIM + NGR + 255) / 256, 256, 0, stream>>>(pooled, NGR * DIM + NGR);
    k_pool_nodes<<<(nN + 7) / 8, 256, 0, stream>>>(bufB, batch, pooled, cnt, nN);
    k_head<<<1, NGR, 0, stream>>>(pooled, cnt, outW, outb, (float*)d_out);
}